// Model_14053132993138
// MI455X (gfx1250) — compile-verified
//
#include <hip/hip_runtime.h>
#include <hip/hip_bf16.h>
#include <cstdint>

typedef __attribute__((ext_vector_type(2))) float v2f;
typedef __attribute__((ext_vector_type(8))) float v8f;

#define ALPHA_P 2.0f
#define BETA_P  1.0f
#define CLIP_P  0.99f

// digamma for x >= 1 (confusion entries are prior(>=1) + nonneg counts)
__device__ __forceinline__ float psi_f(float x) {
    float r = 0.0f;
    while (x < 6.0f) { r -= 1.0f / x; x += 1.0f; }
    float inv  = 1.0f / x;
    float inv2 = inv * inv;
    float series = inv2 * (1.0f/12.0f - inv2 * (1.0f/120.0f - inv2 * (1.0f/252.0f)));
    return r + logf(x) - 0.5f * inv - series;
}

// K0: per-worker digamma message table + (re)initialize outputs each call.
__global__ void k_init(const float* __restrict__ confusion, float* __restrict__ msgTab,
                       float* __restrict__ out, int U) {
    int u = blockIdx.x * blockDim.x + threadIdx.x;
    if (u == 0) out[0] = 0.0f;           // loss accumulator
    if (u >= U) return;
    float c00 = confusion[u*4+0], c01 = confusion[u*4+1];
    float c10 = confusion[u*4+2], c11 = confusion[u*4+3];
    float drs0 = psi_f(c00 + c01);
    float drs1 = psi_f(c10 + c11);
    // msg[u][c][k] = digamma(conf[u][k][c]) - digamma(rowsum_k)
    msgTab[u*4+0] = psi_f(c00) - drs0;   // c=0,k=0
    msgTab[u*4+1] = psi_f(c10) - drs1;   // c=0,k=1
    msgTab[u*4+2] = psi_f(c01) - drs0;   // c=1,k=0
    msgTab[u*4+3] = psi_f(c11) - drs1;   // c=1,k=1
    float* oc = out + 1 + (size_t)u * 4; // new_confusion := prior
    oc[0] = ALPHA_P; oc[1] = BETA_P; oc[2] = BETA_P; oc[3] = ALPHA_P;
}

// K1: logits = x @ W + b via V_WMMA_F32_16X16X4_F32.
// A = W^T zero-padded to 16 rows (documented f32 16x4 layout),
// B = x tile (4 x 16 tasks). One wave -> 16 tasks.
__global__ void k_logits_wmma(const float* __restrict__ x, const float* __restrict__ W,
                              const float* __restrict__ bias, float* __restrict__ logits,
                              int D, int T) {
    extern __shared__ float Wt[];        // [2][D] transposed W
    const int tid = threadIdx.x;
    for (int i = tid; i < D * 2; i += blockDim.x) {
        int c = i / D, d = i - c * D;
        Wt[i] = W[d * 2 + c];
    }
    __syncthreads();

    const int lane  = tid & 31;
    const int gw    = blockIdx.x * (blockDim.x >> 5) + (tid >> 5);
    const int tbase = gw * 16;
    if (tbase + 16 > T) return;          // wave-uniform; no barriers after this

    const int n    = lane & 15;          // A row (class, padded) / B column (task)
    const int half = lane >> 4;          // K sub-block selector
    const float msk = (n < 2) ? 1.0f : 0.0f;            // zero-pad rows 2..15 of A
    const float* xrow = x + (size_t)(tbase + n) * D + 2 * half;
    const float* wrow = Wt + (size_t)(n & 1) * D + 2 * half;

    v8f acc = {};
    #pragma unroll 8
    for (int k = 0; k < D; k += 4) {
        // A 16x4 f32 (doc): lane<16 holds K=k,k+1 ; lane>=16 holds K=k+2,k+3 (row M=lane&15)
        v2f a = *reinterpret_cast<const v2f*>(wrow + k);  // unconditional ds_load_b64
        a[0] *= msk; a[1] *= msk;
        // B 4x16 f32: lane<16 rows K=k,k+1 at N=lane ; lane>=16 rows K=k+2,k+3
        v2f b = *reinterpret_cast<const v2f*>(xrow + k);  // global_load_b64
        acc = __builtin_amdgcn_wmma_f32_16x16x4_f32(false, a, false, b,
                                                    (short)0, acc, false, false);
    }
    // C layout: (M,N) with M<8 -> VGPR M, lane N (lanes 0-15).
    // M=0/1 are the two classes; lane n (<16) holds task tbase+n in acc[0], acc[1].
    if (lane < 16) {
        v2f o;
        o[0] = acc[0] + bias[0];
        o[1] = acc[1] + bias[1];
        *reinterpret_cast<v2f*>(logits + (size_t)(tbase + n) * 2) = o;  // coalesced b64
    }
}

// K2: one wave per task: scan ann row (b128), message sum, softmax/KL, scatter.
__global__ void k_main(const int* __restrict__ ann, const float* __restrict__ logits,
                       const float* __restrict__ msgTab, float* __restrict__ out,
                       int T, int U) {
    __shared__ float sLoss;
    const int tid = threadIdx.x;
    if (tid == 0) sLoss = 0.0f;
    __syncthreads();

    const int lane = tid & 31;
    const int t    = blockIdx.x * (blockDim.x >> 5) + (tid >> 5);
    float myLoss = 0.0f;

    if (t < T) {
        // log(clip(softmax(logits)) + 1e-9)
        float l0 = logits[(size_t)t*2 + 0];
        float l1 = logits[(size_t)t*2 + 1];
        float mx = fmaxf(l0, l1);
        float e0 = expf(l0 - mx), e1 = expf(l1 - mx);
        float ip = 1.0f / (e0 + e1);
        float p0 = fminf(fmaxf(e0 * ip, 1.0f - CLIP_P), CLIP_P) + 1e-9f;
        float p1 = fminf(fmaxf(e1 * ip, 1.0f - CLIP_P), CLIP_P) + 1e-9f;
        float lz0 = logf(p0), lz1 = logf(p1);

        const int* arow = ann + (size_t)t * U;
        float t0 = 0.0f, t1 = 0.0f;
        unsigned rec[8];
        int cnt = 0;

        const int nFull = U >> 7;                       // 128 ints / wave-iter
        const int4* arow4 = reinterpret_cast<const int4*>(arow);
        for (int it = 0; it < nFull; ++it) {
            int4 v = arow4[it * 32 + lane];
            int jb = it * 128 + lane * 4;
            int vv[4] = {v.x, v.y, v.z, v.w};
            #pragma unroll
            for (int c = 0; c < 4; ++c) {
                int a = vv[c];
                if (a >= 0) {
                    int u = jb + c;
                    t0 += msgTab[u*4 + a*2 + 0];
                    t1 += msgTab[u*4 + a*2 + 1];
                    if (cnt < 8) rec[cnt] = (unsigned)((u << 1) | a);
                    ++cnt;
                }
            }
        }
        for (int j = nFull * 128 + lane; j < U; j += 32) {  // tail
            int a = arow[j];
            if (a >= 0) {
                t0 += msgTab[j*4 + a*2 + 0];
                t1 += msgTab[j*4 + a*2 + 1];
                if (cnt < 8) rec[cnt] = (unsigned)((j << 1) | a);
                ++cnt;
            }
        }
        // butterfly reduction -> every lane holds the row totals
        for (int off = 16; off > 0; off >>= 1) {
            t0 += __shfl_xor(t0, off, 32);
            t1 += __shfl_xor(t1, off, 32);
        }
        float s0 = lz0 + t0, s1 = lz1 + t1;
        float ms = fmaxf(s0, s1);
        float f0 = expf(s0 - ms), f1 = expf(s1 - ms);
        float sum = f0 + f1, isum = 1.0f / sum;
        float zi0 = f0 * isum, zi1 = f1 * isum;
        float lse = logf(sum);
        float lzi0 = s0 - ms - lse, lzi1 = s1 - ms - lse;
        if (lane == 0)
            myLoss = zi0 * (lzi0 - lz0) + zi1 * (lzi1 - lz1);

        // scatter z_inference into new_confusion[u][k][ann]
        float* oc = out + 1;
        int kk = cnt < 8 ? cnt : 8;
        for (int i = 0; i < kk; ++i) {
            int u = rec[i] >> 1;
            int a = rec[i] & 1;
            atomicAdd(&oc[u*4 + 0 + a], zi0);   // k=0
            atomicAdd(&oc[u*4 + 2 + a], zi1);   // k=1
        }
    }
    if (lane == 0 && t < T) atomicAdd(&sLoss, myLoss);
    __syncthreads();
    if (tid == 0) atomicAdd(&out[0], sLoss);
}

extern "C" void kernel_launch(void* const* d_in, const int* in_sizes, int n_in,
                              void* d_out, int out_size, void* d_ws, size_t ws_size,
                              hipStream_t stream) {
    (void)n_in; (void)out_size; (void)ws_size;
    const float* x         = (const float*)d_in[0];
    const int*   ann       = (const int*)  d_in[1];
    const float* W         = (const float*)d_in[2];
    const float* b         = (const float*)d_in[3];
    const float* confusion = (const float*)d_in[4];

    const int D = in_sizes[2] / 2;      // 512
    const int T = in_sizes[0] / D;      // 20000
    const int U = in_sizes[4] / 4;      // 2000

    float* out    = (float*)d_out;
    float* logits = (float*)d_ws;             // T*2 floats
    float* msgTab = logits + (size_t)T * 2;   // U*4 floats

    k_init<<<(U + 255) / 256, 256, 0, stream>>>(confusion, msgTab, out, U);

    int nWaves = (T + 15) / 16;
    int blocks = (nWaves + 1) / 2;            // 2 waves (64 threads) per block
    size_t ldsBytes = (size_t)D * 2 * sizeof(float);
    k_logits_wmma<<<blocks, 64, ldsBytes, stream>>>(x, W, b, logits, D, T);

    const int tpb = 256, wpb = tpb / 32;      // 8 tasks per block
    k_main<<<(T + wpb - 1) / wpb, tpb, 0, stream>>>(ann, logits, msgTab, out, T, U);
}